// BiGtNet_63118839382156
// MI455X (gfx1250) — compile-verified
//
#include <hip/hip_runtime.h>
#include <stdint.h>

// ---------------- problem constants ----------------
#define BB 32
#define HH 192
#define WW 192
#define HW (HH * WW)
#define KC 25
#define TAPS 25

// ---------------- tiling ----------------
#define TX 32
#define TY 8
#define HALO_A 2                 // halo where appear/seg is needed
#define HALO_L 4                 // halo where labels are needed
#define LX (TX + 2 * HALO_L)     // 40
#define LY (TY + 2 * HALO_L)     // 16
#define SX (TX + 2 * HALO_A)     // 36
#define SY (TY + 2 * HALO_A)     // 12

// ---------------- output layout (floats) ----------------
#define O_PRED 0
#define O_MASKF (BB * 3 * HW)
#define O_NAPF (O_MASKF + BB * KC * HW)
#define O_ATTN (O_NAPF + BB * HW)
#define O_MASKB (O_ATTN + BB * HW)
#define O_NAPB (O_MASKB + BB * KC * HW)
#define O_NATTN (O_NAPB + BB * HW)

// ---------------- CDNA5 async global->LDS path ----------------
// b32 builtin: (int*, int*, imm, imm); b64 builtin: (v2i*, v2i*, imm, imm).
// clang rewrites pointer address spaces per-arg.
#if defined(__AMDGCN__) && __has_builtin(__builtin_amdgcn_global_load_async_to_lds_b32)
#define USE_ASYNC_LDS 1
#endif
#if defined(__AMDGCN__) && __has_builtin(__builtin_amdgcn_global_load_async_to_lds_b64)
#define HAVE_ASYNC_B64 1
#endif

typedef int b64vec __attribute__((vector_size(8)));

__device__ __forceinline__ void copy_b32_to_lds(const void* g, void* l) {
#ifdef USE_ASYNC_LDS
  __builtin_amdgcn_global_load_async_to_lds_b32((int*)g, (int*)l, 0, 0);
#else
  *(int*)l = *(const int*)g;
#endif
}

__device__ __forceinline__ void copy_b64_to_lds(const void* g, void* l) {
#ifdef HAVE_ASYNC_B64
  __builtin_amdgcn_global_load_async_to_lds_b64((b64vec*)g, (b64vec*)l, 0, 0);
#else
  copy_b32_to_lds(g, l);
  copy_b32_to_lds((const int*)g + 1, (int*)l + 1);
#endif
}

__device__ __forceinline__ void async_lds_wait_all() {
#ifdef USE_ASYNC_LDS
  asm volatile("s_wait_asynccnt 0" ::: "memory");
#endif
}

// ---------------- fused kernel ----------------
__global__ __launch_bounds__(256) void bigtnet_fused(
    const float* __restrict__ im_f, const float* __restrict__ im_b,
    const int* __restrict__ lab_f, const int* __restrict__ lab_b,
    const float* __restrict__ kern, float* __restrict__ out) {
  __shared__ int s_lab[2][LY][LX];          // labels, OOB -> KC (zero row)
  __shared__ float s_wim[2][3][SY][SX];     // im channels, later scaled by appear
  __shared__ float s_seg[2][SY][SX];        // seg on extended region
  __shared__ float s_kt[26 * TAPS];         // tap table, row 25 = zeros

  const int tid = threadIdx.x;
  const int x0 = blockIdx.x * TX;
  const int y0 = blockIdx.y * TY;
  const int b = blockIdx.z;

  // tap table: m_kernel flat layout is exactly k*25 + (ty*5+tx)
  for (int i = tid; i < 26 * TAPS; i += 256) {
    if (i < KC * TAPS)
      copy_b32_to_lds(&kern[i], &s_kt[i]);
    else
      s_kt[i] = 0.0f;
  }

  // labels with halo-4, both directions, as b64 pairs.
  // x0-4 is even and WW is even -> a pair (gx, gx+1) with even gx is either
  // fully in-bounds or fully out-of-bounds (never straddles the border).
  for (int i = tid; i < 2 * LY * (LX / 2); i += 256) {     // 640 pairs
    int dir = i / (LY * (LX / 2));
    int r = i - dir * (LY * (LX / 2));
    int ly = r / (LX / 2), lx = 2 * (r - ly * (LX / 2));
    int gy = y0 - HALO_L + ly, gx = x0 - HALO_L + lx;
    if ((unsigned)gy < (unsigned)HH && (unsigned)gx < (unsigned)WW) {
      const int* src = (dir == 0 ? lab_f : lab_b) + (size_t)b * HW + gy * WW + gx;
      copy_b64_to_lds(src, &s_lab[dir][ly][lx]);
    } else {
      s_lab[dir][ly][lx] = KC;      // zero row of tap table -> zero padding
      s_lab[dir][ly][lx + 1] = KC;
    }
  }

  // image channels 3..5 with halo-2, both directions, as b64 pairs
  for (int i = tid; i < 2 * 3 * SY * (SX / 2); i += 256) { // 1296 pairs
    int dir = i / (3 * SY * (SX / 2));
    int r = i - dir * (3 * SY * (SX / 2));
    int c = r / (SY * (SX / 2));
    r -= c * (SY * (SX / 2));
    int sy = r / (SX / 2), sx = 2 * (r - sy * (SX / 2));
    int gy = y0 - HALO_A + sy, gx = x0 - HALO_A + sx;
    if ((unsigned)gy < (unsigned)HH && (unsigned)gx < (unsigned)WW) {
      const float* src = (dir == 0 ? im_f : im_b) +
                         ((size_t)b * 6 + 3 + c) * HW + gy * WW + gx;
      copy_b64_to_lds(src, &s_wim[dir][c][sy][sx]);
    } else {
      s_wim[dir][c][sy][sx] = 0.0f;
      s_wim[dir][c][sy][sx + 1] = 0.0f;
    }
  }

  async_lds_wait_all();
  __syncthreads();

  // seg over the extended (halo-2) region: 25-tap gather from the tap table
  for (int i = tid; i < 2 * SY * SX; i += 256) {
    int dir = i / (SY * SX);
    int r = i - dir * (SY * SX);
    int sy = r / SX, sx = r - sy * SX;
    float acc = 0.0f;
#pragma unroll
    for (int t = 0; t < TAPS; ++t) {
      int dy = t / 5, dx = t - dy * 5;
      int lab = s_lab[dir][sy + dy][sx + dx];
      acc += s_kt[lab * TAPS + t];
    }
    s_seg[dir][sy][sx] = acc;
  }
  __syncthreads();

  // scale staged image by appear(seg) in place
  for (int i = tid; i < 2 * 3 * SY * SX; i += 256) {
    int dir = i / (3 * SY * SX);
    int r = i - dir * (3 * SY * SX);
    int c = r / (SY * SX);
    r -= c * (SY * SX);
    int sy = r / SX, sx = r - sy * SX;
    float seg = s_seg[dir][sy][sx];
    float ap = fmaxf(1.0f - fmaxf(seg - 1.0f, 0.0f), 0.0f);
    s_wim[dir][c][sy][sx] *= ap;
  }
  __syncthreads();

  // interior pixel per thread: lanes = consecutive x -> coalesced 128B stores
  const int tx = tid & 31, ty = tid >> 5;
  const int px = x0 + tx, py = y0 + ty;
  const size_t pix = (size_t)py * WW + px;

  float segc[2], ap[2], predc[2][3];
#pragma unroll
  for (int dir = 0; dir < 2; ++dir) {
    float s = s_seg[dir][ty + HALO_A][tx + HALO_A];
    segc[dir] = 1.0f - fmaxf(1.0f - s, 0.0f);
    ap[dir] = fmaxf(1.0f - fmaxf(s - 1.0f, 0.0f), 0.0f);
    float p0 = 0.0f, p1 = 0.0f, p2 = 0.0f;
#pragma unroll
    for (int t = 0; t < TAPS; ++t) {
      int dy = t / 5, dx = t - dy * 5;
      int lab = s_lab[dir][ty + HALO_A + dy][tx + HALO_A + dx];
      float w = s_kt[lab * TAPS + t];
      p0 = fmaf(w, s_wim[dir][0][ty + dy][tx + dx], p0);
      p1 = fmaf(w, s_wim[dir][1][ty + dy][tx + dx], p1);
      p2 = fmaf(w, s_wim[dir][2][ty + dy][tx + dx], p2);
    }
    predc[dir][0] = p0;
    predc[dir][1] = p1;
    predc[dir][2] = p2;
  }
  float attn = (segc[0] + 1e-5f) / (segc[0] + segc[1] + 2e-5f);

#pragma unroll
  for (int c = 0; c < 3; ++c) {
    float v = attn * predc[0][c] + (1.0f - attn) * predc[1][c];
    __builtin_nontemporal_store(v, &out[O_PRED + ((size_t)(b * 3 + c)) * HW + pix]);
  }
  __builtin_nontemporal_store(1.0f - ap[0], &out[O_NAPF + (size_t)b * HW + pix]);
  __builtin_nontemporal_store(1.0f - ap[1], &out[O_NAPB + (size_t)b * HW + pix]);
  __builtin_nontemporal_store(attn, &out[O_ATTN + (size_t)b * HW + pix]);
  __builtin_nontemporal_store(1.0f - attn, &out[O_NATTN + (size_t)b * HW + pix]);

  const int lf = s_lab[0][ty + HALO_L][tx + HALO_L];
  const int lb = s_lab[1][ty + HALO_L][tx + HALO_L];
#pragma unroll
  for (int k = 0; k < KC; ++k) {
    __builtin_nontemporal_store((k == lf) ? 1.0f : 0.0f,
                                &out[O_MASKF + ((size_t)(b * KC + k)) * HW + pix]);
    __builtin_nontemporal_store((k == lb) ? 1.0f : 0.0f,
                                &out[O_MASKB + ((size_t)(b * KC + k)) * HW + pix]);
  }
}

extern "C" void kernel_launch(void* const* d_in, const int* in_sizes, int n_in,
                              void* d_out, int out_size, void* d_ws, size_t ws_size,
                              hipStream_t stream) {
  (void)in_sizes; (void)n_in; (void)out_size; (void)d_ws; (void)ws_size;
  const float* im_f = (const float*)d_in[0];
  const float* im_b = (const float*)d_in[1];
  const int* lab_f = (const int*)d_in[2];
  const int* lab_b = (const int*)d_in[3];
  const float* kern = (const float*)d_in[4];
  float* out = (float*)d_out;

  dim3 grid(WW / TX, HH / TY, BB);  // 6 x 24 x 32 = 4608 workgroups
  hipLaunchKernelGGL(bigtnet_fused, grid, dim3(256), 0, stream,
                     im_f, im_b, lab_f, lab_b, kern, out);
}